// SinkhornDistance_75685913690394
// MI455X (gfx1250) — compile-verified
//
#include <hip/hip_runtime.h>
#include <math.h>

#define NB 16
#define P  2048
#define EPSV 0.1f
#define INV_EPS 10.0f

typedef float v2f __attribute__((ext_vector_type(2)));
typedef float v8f __attribute__((ext_vector_type(8)));

// ---------------------------------------------------------------------------
// Kernel A: C[n,i,j] = ||x_i - y_j||  via WMMA f32 16x16x4 Gram + epilogue.
// Block = 256 threads (8 waves). Block tile: 16 rows x 128 cols; wave w owns
// the 16x16 sub-tile at columns [w*16, w*16+16).
// A fragment (16x4, K padded to 4): lanes 0-15 hold (x0,x1), lanes 16-31 (x2,0)
// B fragment (4x16): lanes 0-15 hold (y0,y1), lanes 16-31 (y2,0)
// D layout: lanes 0-15 -> N=lane, M=v; lanes 16-31 -> N=lane-16, M=v+8.
// ---------------------------------------------------------------------------
__global__ __launch_bounds__(256) void cost_matrix_wmma(
    const float* __restrict__ x, const float* __restrict__ y,
    float* __restrict__ Cout) {
  __shared__ float s_x[16][3];
  __shared__ float s_xx[16];
  __shared__ float s_y[128][3];
  __shared__ float s_yy[128];

  const int tid  = threadIdx.x;
  const int n    = blockIdx.z;
  const int row0 = blockIdx.y * 16;
  const int col0 = blockIdx.x * 128;

  if (tid < 16) {
    const float* p = x + ((size_t)n * P + row0 + tid) * 3;
    float a = p[0], b = p[1], c = p[2];
    s_x[tid][0] = a; s_x[tid][1] = b; s_x[tid][2] = c;
    s_xx[tid] = a * a + b * b + c * c;
  }
  if (tid >= 32 && tid < 160) {
    int k = tid - 32;
    const float* p = y + ((size_t)n * P + col0 + k) * 3;
    float a = p[0], b = p[1], c = p[2];
    s_y[k][0] = a; s_y[k][1] = b; s_y[k][2] = c;
    s_yy[k] = a * a + b * b + c * c;
  }
  __syncthreads();

  const int wave = tid >> 5;
  const int lane = tid & 31;
  const int lm   = lane & 15;
  const bool hi  = lane >= 16;

  v2f a;
  a.x = hi ? s_x[lm][2] : s_x[lm][0];
  a.y = hi ? 0.0f       : s_x[lm][1];

  const int cl = wave * 16 + lm;     // column within 128-wide block tile
  v2f b;
  b.x = hi ? s_y[cl][2] : s_y[cl][0];
  b.y = hi ? 0.0f       : s_y[cl][1];

  v8f acc = {};
  acc = __builtin_amdgcn_wmma_f32_16x16x4_f32(
      false, a, false, b, (short)0, acc, false, false);

  const float yy = s_yy[cl];
  const size_t base = ((size_t)n * P + row0) * P + col0 + cl;
#pragma unroll
  for (int vv = 0; vv < 8; ++vv) {
    int r = vv + (hi ? 8 : 0);
    float d2 = s_xx[r] + yy - 2.0f * acc[vv];
    Cout[base + (size_t)r * P] = sqrtf(fmaxf(d2, 0.0f));
  }
}

// ---------------------------------------------------------------------------
// Init: u = v = 0, flag = 0
// ---------------------------------------------------------------------------
__global__ void init_uv(float* __restrict__ uv, int* __restrict__ flag) {
  int k = blockIdx.x * blockDim.x + threadIdx.x;
  if (k < 2 * NB * P) uv[k] = 0.0f;
  if (k == 0) *flag = 0;
}

// ---------------------------------------------------------------------------
// u-update: one wave per row. u_new = eps*log_mu - eps*LSE_j((v_j - C_ij)/eps)
// Online (streaming) stable logsumexp, float4 coalesced loads,
// shfl_xor (m,s)-pair combine. errp[row] = |u_new - u_old|.
// ---------------------------------------------------------------------------
__global__ __launch_bounds__(256) void sink_u(
    const float* __restrict__ C, float* __restrict__ u,
    const float* __restrict__ v, float* __restrict__ errp,
    const int* __restrict__ flag, float log_mu) {
  if (*flag) return;
  const int n    = blockIdx.y;
  const int row  = blockIdx.x * 8 + (threadIdx.x >> 5);
  const int lane = threadIdx.x & 31;
  const float* Crow = C + ((size_t)n * P + row) * P;
  const float* vb   = v + n * P;

  float m = -INFINITY, s = 0.0f;
  for (int j = lane * 4; j < P; j += 128) {
    float4 c4 = *(const float4*)(Crow + j);
    float4 v4 = *(const float4*)(vb + j);
    float t[4] = {(v4.x - c4.x) * INV_EPS, (v4.y - c4.y) * INV_EPS,
                  (v4.z - c4.z) * INV_EPS, (v4.w - c4.w) * INV_EPS};
#pragma unroll
    for (int k = 0; k < 4; ++k) {
      float nm = fmaxf(m, t[k]);
      s = s * __expf(m - nm) + __expf(t[k] - nm);
      m = nm;
    }
  }
#pragma unroll
  for (int off = 16; off > 0; off >>= 1) {
    float mo = __shfl_xor(m, off, 32);
    float so = __shfl_xor(s, off, 32);
    float nm = fmaxf(m, mo);
    s = s * __expf(m - nm) + so * __expf(mo - nm);
    m = nm;
  }
  if (lane == 0) {
    float un  = EPSV * log_mu - EPSV * (m + __logf(s));
    float old = u[n * P + row];
    u[n * P + row]    = un;
    errp[n * P + row] = fabsf(un - old);
  }
}

// ---------------------------------------------------------------------------
// v-update: thread per column. v_new = eps*log_nu - eps*LSE_i((u_i - C_ij)/eps)
// Loads coalesced across threads; u_i is wave-uniform (scalar loads).
// ---------------------------------------------------------------------------
__global__ __launch_bounds__(256) void sink_v(
    const float* __restrict__ C, const float* __restrict__ u,
    float* __restrict__ v, const int* __restrict__ flag, float log_nu) {
  if (*flag) return;
  const int n = blockIdx.y;
  const int j = blockIdx.x * 256 + threadIdx.x;
  const float* Cb = C + (size_t)n * P * P + j;
  const float* ub = u + n * P;

  float m = -INFINITY, s = 0.0f;
#pragma unroll 4
  for (int i = 0; i < P; ++i) {
    float t  = (ub[i] - Cb[(size_t)i * P]) * INV_EPS;
    float nm = fmaxf(m, t);
    s = s * __expf(m - nm) + __expf(t - nm);
    m = nm;
  }
  v[n * P + j] = EPSV * log_nu - EPSV * (m + __logf(s));
}

// ---------------------------------------------------------------------------
// Convergence check: err = (1/NB) * sum over all rows of |du|. Deterministic
// fixed-order tree. Sets flag so all later iteration kernels retire early.
// ---------------------------------------------------------------------------
__global__ __launch_bounds__(256) void sink_check(
    const float* __restrict__ errp, int* __restrict__ flag) {
  __shared__ float red[256];
  if (*flag) return;
  float acc = 0.0f;
  for (int k = threadIdx.x; k < NB * P; k += 256) acc += errp[k];
  red[threadIdx.x] = acc;
  __syncthreads();
#pragma unroll
  for (int off = 128; off > 0; off >>= 1) {
    if (threadIdx.x < off) red[threadIdx.x] += red[threadIdx.x + off];
    __syncthreads();
  }
  if (threadIdx.x == 0) {
    if (red[0] * (1.0f / (float)NB) < 0.1f) *flag = 1;
  }
}

// ---------------------------------------------------------------------------
// Final: pi = exp((u_i + v_j - C)/eps); per-row partial of sum(pi*C).
// ---------------------------------------------------------------------------
__global__ __launch_bounds__(256) void pi_cost(
    const float* __restrict__ C, const float* __restrict__ u,
    const float* __restrict__ v, float* __restrict__ pi,
    float* __restrict__ costp) {
  __shared__ float red[256];
  const int n   = blockIdx.y;
  const int row = blockIdx.x;
  const float un = u[n * P + row];
  const float* vb = v + n * P;
  const size_t base = ((size_t)n * P + row) * P;

  float acc = 0.0f;
  for (int j = threadIdx.x * 4; j < P; j += 1024) {
    float4 c4 = *(const float4*)(C + base + j);
    float4 v4 = *(const float4*)(vb + j);
    float4 p;
    p.x = __expf((un + v4.x - c4.x) * INV_EPS);
    p.y = __expf((un + v4.y - c4.y) * INV_EPS);
    p.z = __expf((un + v4.z - c4.z) * INV_EPS);
    p.w = __expf((un + v4.w - c4.w) * INV_EPS);
    *(float4*)(pi + base + j) = p;
    acc += p.x * c4.x + p.y * c4.y + p.z * c4.z + p.w * c4.w;
  }
  red[threadIdx.x] = acc;
  __syncthreads();
#pragma unroll
  for (int off = 128; off > 0; off >>= 1) {
    if (threadIdx.x < off) red[threadIdx.x] += red[threadIdx.x + off];
    __syncthreads();
  }
  if (threadIdx.x == 0) costp[n * P + row] = red[0];
}

__global__ __launch_bounds__(256) void cost_final(
    const float* __restrict__ costp, float* __restrict__ cost) {
  __shared__ float red[256];
  const int n = blockIdx.x;
  float acc = 0.0f;
  for (int k = threadIdx.x; k < P; k += 256) acc += costp[n * P + k];
  red[threadIdx.x] = acc;
  __syncthreads();
#pragma unroll
  for (int off = 128; off > 0; off >>= 1) {
    if (threadIdx.x < off) red[threadIdx.x] += red[threadIdx.x + off];
    __syncthreads();
  }
  if (threadIdx.x == 0) cost[n] = red[0];
}

// ---------------------------------------------------------------------------
extern "C" void kernel_launch(void* const* d_in, const int* in_sizes, int n_in,
                              void* d_out, int out_size, void* d_ws,
                              size_t ws_size, hipStream_t stream) {
  const float* x = (const float*)d_in[0];
  const float* y = (const float*)d_in[1];

  float* out    = (float*)d_out;
  float* cost_o = out;                               // 16
  float* pi_o   = out + 16;                          // NB*P*P
  float* C_o    = pi_o + (size_t)NB * P * P;         // NB*P*P

  float* wsf   = (float*)d_ws;
  float* u     = wsf;                                // NB*P
  float* v     = u + NB * P;                         // NB*P
  float* errp  = v + NB * P;                         // NB*P
  float* costp = errp + NB * P;                      // NB*P
  int*   flag  = (int*)(costp + NB * P);

  const float log_mu = logf(1.0f / (float)P + 1e-8f);
  const float log_nu = log_mu;

  init_uv<<<dim3((2 * NB * P + 255) / 256), 256, 0, stream>>>(wsf, flag);

  cost_matrix_wmma<<<dim3(P / 128, P / 16, NB), 256, 0, stream>>>(x, y, C_o);

  for (int it = 0; it < 100; ++it) {
    sink_u<<<dim3(P / 8, NB), 256, 0, stream>>>(C_o, u, v, errp, flag, log_mu);
    sink_v<<<dim3(P / 256, NB), 256, 0, stream>>>(C_o, u, v, flag, log_nu);
    sink_check<<<1, 256, 0, stream>>>(errp, flag);
  }

  pi_cost<<<dim3(P, NB), 256, 0, stream>>>(C_o, u, v, pi_o, costp);
  cost_final<<<NB, 256, 0, stream>>>(costp, cost_o);
}